// ContextualAttention_34265249087928
// MI455X (gfx1250) — compile-verified
//
#include <hip/hip_runtime.h>
#include <hip/hip_bf16.h>

typedef float v2f __attribute__((ext_vector_type(2)));
typedef float v8f __attribute__((ext_vector_type(8)));

// Problem constants (from reference)
#define BATCH 4
#define CH    128
#define H0    64
#define W0    64
#define HD    32      // downsampled
#define WD    32
#define L     1024    // HD*WD patches
#define KCOL  1152    // CH*9
#define QDIM  2048    // CH*16
#define SOFTMAX_SCALE 10.0f

// ---------------------------------------------------------------------------
// CDNA5 async global->LDS copy (16B per lane), tracked by ASYNCcnt.
// vdst = LDS byte address (low 32 bits of the flat shared pointer, which is
// exactly what the flat-aperture truncation uses), vaddr = 64-bit global addr.
// ---------------------------------------------------------------------------
__device__ __forceinline__ void async_copy16(unsigned lds_byte_off, const float* g) {
    asm volatile("global_load_async_to_lds_b128 %0, %1, off"
                 :: "v"(lds_byte_off), "v"((unsigned long long)(uintptr_t)g)
                 : "memory");
}
__device__ __forceinline__ void wait_async0() {
    asm volatile("s_wait_asynccnt 0x0" ::: "memory");
}
__device__ __forceinline__ unsigned lds_addr32(const void* p) {
    return (unsigned)(uintptr_t)p;
}

// ---------------------------------------------------------------------------
// im2col of the downsampled (::2) tensor with 3x3 SAME patches.
// col[b][p][c*9+t] = x[b, c, 2*(py+dy-1), 2*(px+dx-1)] (zero padded)
// ---------------------------------------------------------------------------
__global__ void im2col_ds_k3(const float* __restrict__ x, float* __restrict__ col,
                             int total) {
    int idx = blockIdx.x * blockDim.x + threadIdx.x;
    if (idx >= total) return;
    int k = idx % KCOL;
    int p = (idx / KCOL) % L;
    int b = idx / (KCOL * L);
    int c = k / 9, t = k % 9;
    int dy = t / 3, dx = t % 3;
    int py = p >> 5, px = p & 31;
    int y = py + dy - 1, xx = px + dx - 1;
    float v = 0.0f;
    if (y >= 0 && y < HD && xx >= 0 && xx < WD)
        v = x[(((size_t)b * CH + c) * H0 + 2 * y) * W0 + 2 * xx];
    col[idx] = v;
}

// ---------------------------------------------------------------------------
// rawW^T[b][q][p] with q = c*16 + a*4 + bb :
//   raw_w[p,c,a,bb] = b_full[b, c, 2*py + a - 1, 2*px + bb - 1] (zero padded)
// ---------------------------------------------------------------------------
__global__ void build_rawWT(const float* __restrict__ bsrc, float* __restrict__ rawWT,
                            int total) {
    int idx = blockIdx.x * blockDim.x + threadIdx.x;
    if (idx >= total) return;
    int p = idx % L;
    int q = (idx / L) % QDIM;
    int b = idx / (L * QDIM);
    int c = q >> 4, a = (q >> 2) & 3, bb = q & 3;
    int py = p >> 5, px = p & 31;
    int Y = 2 * py + a - 1, X = 2 * px + bb - 1;
    float v = 0.0f;
    if (Y >= 0 && Y < H0 && X >= 0 && X < W0)
        v = bsrc[(((size_t)b * CH + c) * H0 + Y) * W0 + X];
    rawWT[idx] = v;
}

// ---------------------------------------------------------------------------
// rnorm[b][p] = 1 / max(sqrt(sum_k colB[b][p][k]^2), 1e-4)
// ---------------------------------------------------------------------------
__global__ void row_rnorm(const float* __restrict__ colB, float* __restrict__ rnorm) {
    __shared__ float red[128];
    int row = blockIdx.x;                 // b*L + p
    const float* r = colB + (size_t)row * KCOL;
    float s = 0.0f;
    for (int k = threadIdx.x; k < KCOL; k += 128) { float v = r[k]; s += v * v; }
    red[threadIdx.x] = s;
    __syncthreads();
    for (int st = 64; st > 0; st >>= 1) {
        if (threadIdx.x < st) red[threadIdx.x] += red[threadIdx.x + st];
        __syncthreads();
    }
    if (threadIdx.x == 0)
        rnorm[row] = 1.0f / fmaxf(sqrtf(red[0]), 1e-4f);
}

// ---------------------------------------------------------------------------
// mm[p] = 1 if all 9 taps of the (zero-padded) 3x3 patch of mask[0,0,::8,::8]
// are zero, else 0.
// ---------------------------------------------------------------------------
__global__ void build_mm(const float* __restrict__ mask, float* __restrict__ mm) {
    int p = blockIdx.x * blockDim.x + threadIdx.x;
    if (p >= L) return;
    int py = p >> 5, px = p & 31;
    float s = 0.0f;
    for (int dy = 0; dy < 3; ++dy)
        for (int dx = 0; dx < 3; ++dx) {
            int y = py + dy - 1, x = px + dx - 1;
            if (y >= 0 && y < 32 && x >= 0 && x < 32)
                s += mask[(size_t)(8 * y) * 256 + 8 * x];
        }
    mm[p] = (s == 0.0f) ? 1.0f : 0.0f;
}

// ---------------------------------------------------------------------------
// NT GEMM with f32 WMMA:  C[m,n] = (Σ_k A[m,k]*B[n,k]) * rowScale[m]
// A: M x K row-major, B: N x K row-major, C: M x N row-major.
// 128 threads = 4 waves (2x2), block tile 64x64, BK=16, double-buffered LDS
// filled with CDNA5 async global->LDS DMA (ASYNCcnt).
// ---------------------------------------------------------------------------
#define LDS_STRIDE 20
#define TILE_F     (64 * LDS_STRIDE)      // floats per 64x16 padded tile
__global__ __launch_bounds__(128)
void gemm_nt_wmma(const float* __restrict__ A, const float* __restrict__ B,
                  float* __restrict__ C, int M, int N, int K,
                  long strideA, long strideB, long strideC,
                  const float* __restrict__ rowScale, int scaleStride) {
    const int batch = blockIdx.z;
    A += (size_t)batch * strideA;
    B += (size_t)batch * strideB;
    C += (size_t)batch * strideC;
    if (rowScale) rowScale += (size_t)batch * scaleStride;

    const int tid  = threadIdx.x;
    const int lane = tid & 31;
    const int wave = tid >> 5;
    const int wr = wave & 1;          // wave row  (2 waves along M)
    const int wc = wave >> 1;         // wave col  (2 waves along N)
    const int hf = lane >> 4;         // lane half (K-pair select)
    const int lx = lane & 15;

    const int blockM = blockIdx.y * 64;
    const int blockN = blockIdx.x * 64;

    // [buf][A|B][64*20] double-buffered tiles
    __shared__ float smem[2 * 2 * TILE_F];

    v8f acc00 = 0.0f, acc01 = 0.0f, acc10 = 0.0f, acc11 = 0.0f;

    // cooperative-load assignment: thread -> rows lr, lr+32; float4 at col lc
    const int lr = tid >> 2;          // 0..31
    const int lc = (tid & 3) << 2;    // 0,4,8,12

    const float* gA0 = A + (size_t)(blockM + lr)      * K + lc;
    const float* gA1 = A + (size_t)(blockM + lr + 32) * K + lc;
    const float* gB0 = B + (size_t)(blockN + lr)      * K + lc;
    const float* gB1 = B + (size_t)(blockN + lr + 32) * K + lc;

    unsigned dA0[2], dA1[2], dB0[2], dB1[2];
    #pragma unroll
    for (int bf = 0; bf < 2; ++bf) {
        float* As = &smem[bf * 2 * TILE_F];
        float* Bs = As + TILE_F;
        dA0[bf] = lds_addr32(&As[lr * LDS_STRIDE + lc]);
        dA1[bf] = lds_addr32(&As[(lr + 32) * LDS_STRIDE + lc]);
        dB0[bf] = lds_addr32(&Bs[lr * LDS_STRIDE + lc]);
        dB1[bf] = lds_addr32(&Bs[(lr + 32) * LDS_STRIDE + lc]);
    }

    // prologue: fill buffer 0 with tile k0=0
    async_copy16(dA0[0], gA0);
    async_copy16(dA1[0], gA1);
    async_copy16(dB0[0], gB0);
    async_copy16(dB1[0], gB1);
    wait_async0();
    __syncthreads();

    const int nsteps = K >> 4;
    for (int s = 0; s < nsteps; ++s) {
        const int cur = s & 1;
        const int nxt = cur ^ 1;

        // issue async DMA for the next tile while we compute this one
        if (s + 1 < nsteps) {
            const size_t ko = (size_t)(s + 1) << 4;
            async_copy16(dA0[nxt], gA0 + ko);
            async_copy16(dA1[nxt], gA1 + ko);
            async_copy16(dB0[nxt], gB0 + ko);
            async_copy16(dB1[nxt], gB1 + ko);
        }

        const float* As = &smem[cur * 2 * TILE_F];
        const float* Bs = As + TILE_F;

        #pragma unroll
        for (int ks = 0; ks < 4; ++ks) {
            const int kb = ks * 4 + 2 * hf;   // even -> 8B-aligned v2f
            v2f a0 = *(const v2f*)&As[(wr * 32 +  0 + lx) * LDS_STRIDE + kb];
            v2f a1 = *(const v2f*)&As[(wr * 32 + 16 + lx) * LDS_STRIDE + kb];
            v2f b0 = *(const v2f*)&Bs[(wc * 32 +  0 + lx) * LDS_STRIDE + kb];
            v2f b1 = *(const v2f*)&Bs[(wc * 32 + 16 + lx) * LDS_STRIDE + kb];
            acc00 = __builtin_amdgcn_wmma_f32_16x16x4_f32(false, a0, false, b0,
                        (short)0, acc00, false, false);
            acc01 = __builtin_amdgcn_wmma_f32_16x16x4_f32(false, a0, false, b1,
                        (short)0, acc01, false, false);
            acc10 = __builtin_amdgcn_wmma_f32_16x16x4_f32(false, a1, false, b0,
                        (short)0, acc10, false, false);
            acc11 = __builtin_amdgcn_wmma_f32_16x16x4_f32(false, a1, false, b1,
                        (short)0, acc11, false, false);
        }

        wait_async0();       // my next-tile DMA landed
        __syncthreads();     // everyone's landed; everyone done reading cur
    }

    // Epilogue. C/D layout: VGPR r holds M = r (lanes 0-15) / r+8 (lanes 16-31),
    // N = lane & 15.
    #pragma unroll
    for (int mt = 0; mt < 2; ++mt) {
        v8f ar0 = (mt == 0) ? acc00 : acc10;
        v8f ar1 = (mt == 0) ? acc01 : acc11;
        #pragma unroll
        for (int r = 0; r < 8; ++r) {
            int gm = blockM + wr * 32 + mt * 16 + r + 8 * hf;
            float sc = rowScale ? rowScale[gm] : 1.0f;
            int gn0 = blockN + wc * 32 + lx;
            C[(size_t)gm * N + gn0]      = ar0[r] * sc;
            C[(size_t)gm * N + gn0 + 16] = ar1[r] * sc;
        }
    }
}

// ---------------------------------------------------------------------------
// fuse1: out[i,j] = sum_{d=-1..1} in[i+d, j+d]  (flattened 1024x1024, zero OOB)
// ---------------------------------------------------------------------------
__global__ void fuse1(const float* __restrict__ in, float* __restrict__ out, int total) {
    int idx = blockIdx.x * blockDim.x + threadIdx.x;
    if (idx >= total) return;
    int j = idx % L;
    int i = (idx / L) % L;
    int b = idx / (L * L);
    const float* src = in + (size_t)b * L * L;
    float acc = 0.0f;
    #pragma unroll
    for (int d = -1; d <= 1; ++d) {
        int ii = i + d, jj = j + d;
        if (ii >= 0 && ii < L && jj >= 0 && jj < L)
            acc += src[(size_t)ii * L + jj];
    }
    out[idx] = acc;
}

// ---------------------------------------------------------------------------
// fuse2: fuse along the transposed flattening. For p=(py,px), s=(fy,fx):
//   i' = px*32+py, j' = fx*32+fy ; out += in[p(t), s(u)] for t=i'+d, u=j'+d
//   with p(t) = (t%32)*32 + t/32.
// ---------------------------------------------------------------------------
__global__ void fuse2(const float* __restrict__ in, float* __restrict__ out, int total) {
    int idx = blockIdx.x * blockDim.x + threadIdx.x;
    if (idx >= total) return;
    int s = idx % L;
    int p = (idx / L) % L;
    int b = idx / (L * L);
    const float* src = in + (size_t)b * L * L;
    int py = p >> 5, px = p & 31;
    int fy = s >> 5, fx = s & 31;
    int ip = px * 32 + py;
    int jp = fx * 32 + fy;
    float acc = 0.0f;
    #pragma unroll
    for (int d = -1; d <= 1; ++d) {
        int t = ip + d, u = jp + d;
        if (t >= 0 && t < L && u >= 0 && u < L) {
            int psrc = (t & 31) * 32 + (t >> 5);
            int ssrc = (u & 31) * 32 + (u >> 5);
            acc += src[(size_t)psrc * L + ssrc];
        }
    }
    out[idx] = acc;
}

// ---------------------------------------------------------------------------
// Masked softmax over p (axis 0) for each column s, writing transposed:
//   yiT[b][s][p] = softmax_p(in[b][p][s]*mm[p]*10) * mm[p]
// ---------------------------------------------------------------------------
__global__ void softmax_T(const float* __restrict__ in, const float* __restrict__ mm,
                          float* __restrict__ yiT) {
    __shared__ float red[256];
    int s = blockIdx.x;
    int b = blockIdx.y;
    const float* src = in + (size_t)b * L * L;
    float v[4], m[4];
    #pragma unroll
    for (int k = 0; k < 4; ++k) {
        int p = threadIdx.x + k * 256;
        m[k] = mm[p];
        v[k] = src[(size_t)p * L + s] * m[k] * SOFTMAX_SCALE;
    }
    float mx = fmaxf(fmaxf(v[0], v[1]), fmaxf(v[2], v[3]));
    red[threadIdx.x] = mx;
    __syncthreads();
    for (int st = 128; st > 0; st >>= 1) {
        if (threadIdx.x < st) red[threadIdx.x] = fmaxf(red[threadIdx.x], red[threadIdx.x + st]);
        __syncthreads();
    }
    mx = red[0];
    __syncthreads();
    float e[4], ssum = 0.0f;
    #pragma unroll
    for (int k = 0; k < 4; ++k) { e[k] = expf(v[k] - mx); ssum += e[k]; }
    red[threadIdx.x] = ssum;
    __syncthreads();
    for (int st = 128; st > 0; st >>= 1) {
        if (threadIdx.x < st) red[threadIdx.x] += red[threadIdx.x + st];
        __syncthreads();
    }
    float inv = 1.0f / red[0];
    float* dst = yiT + ((size_t)b * L + s) * L;
    #pragma unroll
    for (int k = 0; k < 4; ++k) {
        int p = threadIdx.x + k * 256;
        dst[p] = e[k] * inv * m[k];
    }
}

// ---------------------------------------------------------------------------
// col2im for the stride-2 transposed conv with 4x4 kernel, pad 2:
// out[b,c,Y,X] = 0.25 * Σ_{ky,kx: (Y+ky-2) even,in-range} T[b][(y,x)][c*16+(3-ky)*4+(3-kx)]
// ---------------------------------------------------------------------------
__global__ void col2im_deconv(const float* __restrict__ T, float* __restrict__ out,
                              int total) {
    int idx = blockIdx.x * blockDim.x + threadIdx.x;
    if (idx >= total) return;
    int X = idx & 63;
    int Y = (idx >> 6) & 63;
    int c = (idx >> 12) & 127;
    int b = idx >> 19;
    const float* Tb = T + (size_t)b * L * QDIM;
    float acc = 0.0f;
    #pragma unroll
    for (int ky = 0; ky < 4; ++ky) {
        int u = Y + ky - 2;
        if (u < 0 || u > 62 || (u & 1)) continue;
        int y = u >> 1;
        int a = 3 - ky;
        #pragma unroll
        for (int kx = 0; kx < 4; ++kx) {
            int v = X + kx - 2;
            if (v < 0 || v > 62 || (v & 1)) continue;
            int x = v >> 1;
            int bb = 3 - kx;
            acc += Tb[(size_t)(y * 32 + x) * QDIM + c * 16 + a * 4 + bb];
        }
    }
    out[idx] = acc * 0.25f;
}

// ---------------------------------------------------------------------------
extern "C" void kernel_launch(void* const* d_in, const int* in_sizes, int n_in,
                              void* d_out, int out_size, void* d_ws, size_t ws_size,
                              hipStream_t stream) {
    const float* f    = (const float*)d_in[0];   // (4,128,64,64)
    const float* bsrc = (const float*)d_in[1];   // (4,128,64,64)
    const float* mask = (const float*)d_in[2];   // (4,1,256,256)
    float* out = (float*)d_out;

    // workspace layout (bytes)
    const size_t SZ_COL  = (size_t)BATCH * L * KCOL * 4;   // 18,874,368
    const size_t SZ_YI   = (size_t)BATCH * L * L * 4;      // 16,777,216
    const size_t SZ_RAW  = (size_t)BATCH * QDIM * L * 4;   // 33,554,432
    char* w = (char*)d_ws;
    float* colB   = (float*)(w);
    float* colF   = (float*)(w + SZ_COL);
    float* yiA    = (float*)(w + 2 * SZ_COL);
    float* yiB    = (float*)(w + 2 * SZ_COL + SZ_YI);
    float* rawWT  = (float*)(w + 2 * SZ_COL + 2 * SZ_YI);
    float* Tbuf   = (float*)(w + 2 * SZ_COL + 2 * SZ_YI + SZ_RAW);
    float* rnorm  = (float*)(w + 2 * SZ_COL + 2 * SZ_YI + 2 * SZ_RAW);
    float* mmbuf  = rnorm + BATCH * L;

    const int totCol = BATCH * L * KCOL;
    im2col_ds_k3<<<(totCol + 255) / 256, 256, 0, stream>>>(bsrc, colB, totCol);
    im2col_ds_k3<<<(totCol + 255) / 256, 256, 0, stream>>>(f, colF, totCol);

    const int totRaw = BATCH * QDIM * L;
    build_rawWT<<<(totRaw + 255) / 256, 256, 0, stream>>>(bsrc, rawWT, totRaw);

    row_rnorm<<<BATCH * L, 128, 0, stream>>>(colB, rnorm);
    build_mm<<<(L + 255) / 256, 256, 0, stream>>>(mask, mmbuf);

    // GEMM1: yiA[b][p][s] = rnorm[p] * Σ_k colB[p,k] colF[s,k]   (1024x1024x1152)
    {
        dim3 grid(L / 64, L / 64, BATCH);
        gemm_nt_wmma<<<grid, 128, 0, stream>>>(colB, colF, yiA, L, L, KCOL,
                                               (long)L * KCOL, (long)L * KCOL,
                                               (long)L * L, rnorm, L);
    }

    const int totYi = BATCH * L * L;
    fuse1<<<(totYi + 255) / 256, 256, 0, stream>>>(yiA, yiB, totYi);
    fuse2<<<(totYi + 255) / 256, 256, 0, stream>>>(yiB, yiA, totYi);

    {
        dim3 grid(L, BATCH);
        softmax_T<<<grid, 256, 0, stream>>>(yiA, mmbuf, yiB);   // yiB = yiT (s,p)
    }

    // GEMM2: T[b][s][q] = Σ_p yiT[s,p] * rawWT[q,p]   (1024x2048x1024)
    {
        dim3 grid(QDIM / 64, L / 64, BATCH);
        gemm_nt_wmma<<<grid, 128, 0, stream>>>(yiB, rawWT, Tbuf, L, QDIM, L,
                                               (long)L * L, (long)QDIM * L,
                                               (long)L * QDIM, nullptr, 0);
    }

    const int totOut = BATCH * CH * H0 * W0;
    col2im_deconv<<<(totOut + 255) / 256, 256, 0, stream>>>(Tbuf, out, totOut);
}